// SoftAttentionWeight_9_1__89713276879305
// MI455X (gfx1250) — compile-verified
//
#include <hip/hip_runtime.h>
#include <math.h>

// Problem constants (from the reference)
#define AG    32      // agents per graph
#define ACTD  64      // action dim
#define IND   128     // in dim
#define OUTD  128     // out dim (d_k)
#define NB    256     // graphs
#define NNODE (NB*AG) // 8192 nodes
#define CHOUT 192     // 128 obs + 64 act channels

typedef float v2f __attribute__((ext_vector_type(2)));
typedef float v8f __attribute__((ext_vector_type(8)));

// ---------------------------------------------------------------------------
// Kernel 1: fused per-graph  K/Q projection (WMMA f32 16x16x4)  +  q.k^T
//           attention (WMMA)  +  softmax  ->  w written to d_out w-region.
// One block (4 wave32) per graph.
// ---------------------------------------------------------------------------
__global__ __launch_bounds__(128)
void kq_attn_kernel(const float* __restrict__ h,
                    const float* __restrict__ Wk, const float* __restrict__ bk,
                    const float* __restrict__ Wq, const float* __restrict__ bq,
                    float* __restrict__ w_out)   // [NB*AG*AG]
{
    __shared__ float lh[AG][IND];    // 16 KB
    __shared__ float lk[AG][OUTD];   // 16 KB
    __shared__ float lq[AG][OUTD];   // 16 KB
    __shared__ float ls[AG][AG];     // 4 KB  scores

    const int b    = blockIdx.x;
    const int tid  = threadIdx.x;
    const int lane = tid & 31;
    const int wave = tid >> 5;
    const int half = lane >> 4;   // which 16-lane half of the wave
    const int l16  = lane & 15;

    // Stage h_b (32x128) into LDS, coalesced.
    for (int idx = tid; idx < AG * IND; idx += 128) {
        lh[idx >> 7][idx & 127] = h[(size_t)(b * AG + (idx >> 7)) * IND + (idx & 127)];
    }
    __syncthreads();

    // ---- K and Q projections: 32 output tiles of 16x16 (16 for K, 16 for Q),
    //      8 tiles per wave, K-loop of 32 x V_WMMA_F32_16X16X4_F32 each.
    for (int t = wave; t < 32; t += 4) {
        const bool  isq  = (t >= 16);
        const int   tt   = t & 15;
        const int   mt   = tt >> 3;        // 0..1  (row tile of 32 rows)
        const int   nt   = tt & 7;         // 0..7  (col tile of 128 cols)
        const float* W    = isq ? Wq : Wk;
        const float* bias = isq ? bq : bk;
        const int row = mt * 16 + l16;     // A-frag row for this lane
        const int col = nt * 16 + l16;     // B/C-frag column for this lane

        v8f acc = {0.f, 0.f, 0.f, 0.f, 0.f, 0.f, 0.f, 0.f};
        for (int kk = 0; kk < IND; kk += 4) {
            const int ka = kk + 2 * half;              // ISA A 16x4 layout
            v2f a;  a.x = lh[row][ka];         a.y = lh[row][ka + 1];
            v2f bb; bb.x = W[ka * OUTD + col]; bb.y = W[(ka + 1) * OUTD + col];
            acc = __builtin_amdgcn_wmma_f32_16x16x4_f32(
                      false, a, false, bb, (short)0, acc, false, false);
        }
        const float bv  = bias[col];
        float* dst = isq ? &lq[0][0] : &lk[0][0];
        #pragma unroll
        for (int r = 0; r < 8; ++r) {                  // C/D layout: M = r + 8*half
            dst[(mt * 16 + r + 8 * half) * OUTD + col] = acc[r] + bv;
        }
    }
    __syncthreads();

    // ---- scores = q . k^T / sqrt(128): 4 tiles of 16x16, one per wave.
    {
        const int mt = wave >> 1;   // dst-agent tile
        const int nt = wave & 1;    // src-agent tile
        v8f acc = {0.f, 0.f, 0.f, 0.f, 0.f, 0.f, 0.f, 0.f};
        for (int kk = 0; kk < OUTD; kk += 4) {
            const int ka = kk + 2 * half;
            v2f a;  a.x = lq[mt * 16 + l16][ka];  a.y = lq[mt * 16 + l16][ka + 1];
            // B[d][j] = k[j][d]  -> read k rows directly, no transpose needed
            v2f bb; bb.x = lk[nt * 16 + l16][ka]; bb.y = lk[nt * 16 + l16][ka + 1];
            acc = __builtin_amdgcn_wmma_f32_16x16x4_f32(
                      false, a, false, bb, (short)0, acc, false, false);
        }
        const float sc = 0.08838834764831845f;  // 1/sqrt(128)
        #pragma unroll
        for (int r = 0; r < 8; ++r) {
            ls[mt * 16 + r + 8 * half][nt * 16 + l16] = acc[r] * sc;
        }
    }
    __syncthreads();

    // ---- softmax over src dim (32), one lane per dst agent; write w.
    if (tid < AG) {
        const int i = tid;
        float m = -3.402823466e38f;
        for (int j = 0; j < AG; ++j) m = fmaxf(m, ls[i][j]);
        float s = 0.f;
        for (int j = 0; j < AG; ++j) {
            const float e = expf(ls[i][j] - m);
            ls[i][j] = e;
            s += e;
        }
        const float inv = 1.0f / s;
        for (int j = 0; j < AG; ++j) {
            w_out[(size_t)b * (AG * AG) + i * AG + j] = ls[i][j] * inv;
        }
    }
}

// ---------------------------------------------------------------------------
// Kernel 2: bandwidth stage. One block per (b,i) node.
//   z[j][c]   = pi[j][c] + w[j]*(act[j][c]-pi[j][c]) + noise[b,i,j,c]
//   zbar[c]   = mean_j z[j][c]
//   out[t,ch] = ch<128 ? obs_proc[b,t,ch] : zbar + (pi[t]-z[t])/32
// ---------------------------------------------------------------------------
__global__ __launch_bounds__(256)
void zmix_kernel(const float* __restrict__ policies,
                 const float* __restrict__ actions,
                 const float* __restrict__ obs_proc,
                 const float* __restrict__ noise,
                 const float* __restrict__ w,      // [NNODE*AG]
                 float* __restrict__ out)          // [NNODE*AG*CHOUT]
{
    __shared__ float sw[AG];
    __shared__ float sz[AG][ACTD];   // 8 KB
    __shared__ float szb[ACTD];

    const int node = blockIdx.x;
    const int b    = node >> 5;
    const int tid  = threadIdx.x;

    if (tid < AG) sw[tid] = w[(size_t)node * AG + tid];
    __syncthreads();

    // Materialize z: 32x64 terms, 8 per thread, coalesced noise reads.
    const int c  = tid & 63;
    const int jb = tid >> 6;   // 0..3
    #pragma unroll
    for (int m = 0; m < 8; ++m) {
        const int j  = jb * 8 + m;
        const float wv = sw[j];
        const float p  = policies[(size_t)(b * AG + j) * ACTD + c];
        const float a  = actions [(size_t)(b * AG + j) * ACTD + c];
        sz[j][c] = p + wv * (a - p) + noise[(size_t)node * (AG * ACTD) + j * ACTD + c];
    }
    __syncthreads();

    if (tid < ACTD) {
        float s = 0.f;
        #pragma unroll
        for (int j = 0; j < AG; ++j) s += sz[j][tid];
        szb[tid] = s * (1.0f / 32.0f);
    }
    __syncthreads();

    // Stream out 32*192 floats for this node.
    const size_t base = (size_t)node * (AG * CHOUT);
    for (int e = tid; e < AG * CHOUT; e += 256) {
        const int t  = e / CHOUT;
        const int ch = e - t * CHOUT;
        float v;
        if (ch < IND) {
            v = obs_proc[(size_t)(b * AG + t) * IND + ch];
        } else {
            const int c2 = ch - IND;
            v = szb[c2] + (policies[(size_t)(b * AG + t) * ACTD + c2] - sz[t][c2])
                          * (1.0f / 32.0f);
        }
        out[base + e] = v;
    }
}

// ---------------------------------------------------------------------------
extern "C" void kernel_launch(void* const* d_in, const int* in_sizes, int n_in,
                              void* d_out, int out_size, void* d_ws, size_t ws_size,
                              hipStream_t stream)
{
    const float* h        = (const float*)d_in[0];
    const float* policies = (const float*)d_in[1];
    const float* actions  = (const float*)d_in[2];
    const float* obs_proc = (const float*)d_in[3];
    const float* noise    = (const float*)d_in[4];
    const float* Wk       = (const float*)d_in[5];
    const float* bk       = (const float*)d_in[6];
    const float* Wq       = (const float*)d_in[7];
    const float* bq       = (const float*)d_in[8];

    float* out = (float*)d_out;
    const size_t OBS_SIZE = (size_t)NB * AG * AG * CHOUT;  // 50,331,648 floats
    float* w_out = out + OBS_SIZE;                         // 262,144 floats

    // Stage 1+2 fused: projections + attention + softmax (WMMA f32).
    kq_attn_kernel<<<NB, 128, 0, stream>>>(h, Wk, bk, Wq, bq, w_out);
    // Stage 3: bandwidth-bound mix/concat, one block per (graph, dst agent).
    zmix_kernel<<<NNODE, 256, 0, stream>>>(policies, actions, obs_proc, noise,
                                           w_out, out);
}